// DirectPredictionGNN_12317966205313
// MI455X (gfx1250) — compile-verified
//
#include <hip/hip_runtime.h>
#include <hip/hip_bf16.h>

// ---------------------------------------------------------------------------
// DirectPredictionGNN for MI455X (gfx1250, wave32).
// GEMMs use V_WMMA_F32_16X16X4_F32 (full fp32 precision, matrix pipe).
// Edge aggregation uses hardware global_atomic_add_f32 into an accumulator
// that lives in the 192MB L2 working set.
// ---------------------------------------------------------------------------

#define H 128
#define NLAYERS 4

typedef __attribute__((ext_vector_type(2))) float v2f;
typedef __attribute__((ext_vector_type(8))) float v8f;

__device__ __forceinline__ void atomic_add_f32_global(float* p, float v) {
  // Hardware f32 atomic add (no return) -> STOREcnt; avoids CAS-loop fallback.
  asm volatile("global_atomic_add_f32 %0, %1, off" :: "v"(p), "v"(v) : "memory");
}

// --------------------------- degree / norm ---------------------------------
__global__ void init_deg_kernel(float* __restrict__ deg, int N) {
  int i = blockIdx.x * blockDim.x + threadIdx.x;
  if (i < N) deg[i] = 1.0f;  // self-loop contribution
}

__global__ void deg_accum_kernel(const int* __restrict__ dst, float* __restrict__ deg, int E) {
  int e = blockIdx.x * blockDim.x + threadIdx.x;
  if (e < E) atomic_add_f32_global(deg + dst[e], 1.0f);
}

__global__ void dinv_kernel(const float* __restrict__ deg, float* __restrict__ dinv, int N) {
  int i = blockIdx.x * blockDim.x + threadIdx.x;
  if (i < N) dinv[i] = rsqrtf(deg[i]);  // deg >= 1 always (self loops)
}

// --------------------------- WMMA GEMM -------------------------------------
// C[M,128] = A[M,K] @ W[K,128] (+ bias), K in {32,128}, row-major inputs.
// Block: 256 threads = 8 waves; wave w computes rows [blk*128 + w*16, +16).
// W is staged TRANSPOSED in LDS (sWT[col*K + k]) so each B fragment is a
// single contiguous 8B ds_load_b64 into an even VGPR pair (no reg shuffles).
// Out-of-range A rows are clamped (D rows are independent) and dropped at
// the store, keeping the K-loop branch-free with straight-line b64 loads.
//
// V_WMMA_F32_16X16X4_F32 operand layouts (wave32):
//   A 16x4 : lanes 0-15 -> M=lane, regs {K=0,K=1}; lanes 16-31 -> M=lane-16, {K=2,K=3}
//   B 4x16 : lanes 0-15 -> N=lane, regs {K=0,K=1}; lanes 16-31 -> N=lane-16, {K=2,K=3}
//   C 16x16: lanes 0-15 -> N=lane, reg r -> M=r;   lanes 16-31 -> N=lane-16, reg r -> M=r+8
template <int K>
__global__ __launch_bounds__(256)
void gemm_wmma_kernel(const float* __restrict__ A, const float* __restrict__ W,
                      const float* __restrict__ bias, float* __restrict__ C,
                      int M) {
  __shared__ float sWT[H * K];  // <= 64KB (K=128), column-major: sWT[c*K + k]
  const int tid = threadIdx.x;
  for (int idx = tid; idx < K * H; idx += 256) {
    const int k = idx >> 7;      // idx / 128  (coalesced global read)
    const int c = idx & (H - 1);
    sWT[c * K + k] = W[idx];     // one-time bank-conflicted transpose store
  }
  __syncthreads();

  const int wave  = tid >> 5;
  const int lane  = tid & 31;
  const int mlane = lane & 15;
  const int khalf = (lane >> 4) << 1;           // 0 or 2
  const int row0  = (blockIdx.x * 8 + wave) * 16;
  int arow = row0 + mlane;
  if (arow >= M) arow = M - 1;                  // clamp: branch-free A loads
  const float* aptr = A + (size_t)arow * K + khalf;

  v8f acc[8] = {};

#pragma unroll 2
  for (int k0 = 0; k0 < K; k0 += 4) {
    const v2f a = *(const v2f*)(aptr + k0);     // global_load_b64, 8B aligned
#pragma unroll
    for (int t = 0; t < 8; ++t) {
      const int col = t * 16 + mlane;
      const v2f b = *(const v2f*)(&sWT[col * K + k0 + khalf]);  // ds_load_b64
      acc[t] = __builtin_amdgcn_wmma_f32_16x16x4_f32(
          /*neg_a=*/false, a, /*neg_b=*/false, b,
          /*c_mod=*/(short)0, acc[t], /*reuse_a=*/false, /*reuse_b=*/false);
    }
  }

  // Store D: lane<16 -> rows row0+r, lane>=16 -> rows row0+8+r; col = t*16+mlane.
  const int crow0 = row0 + ((lane >> 4) << 3);
  if (row0 + 16 <= M) {  // uniform fast path: whole tile in range
#pragma unroll
    for (int t = 0; t < 8; ++t) {
      const int col = t * 16 + mlane;
      const float badd = bias ? bias[col] : 0.0f;
#pragma unroll
      for (int r = 0; r < 8; ++r)
        C[(size_t)(crow0 + r) * H + col] = acc[t][r] + badd;
    }
  } else {               // guarded tail tile
#pragma unroll
    for (int t = 0; t < 8; ++t) {
      const int col = t * 16 + mlane;
      const float badd = bias ? bias[col] : 0.0f;
#pragma unroll
      for (int r = 0; r < 8; ++r) {
        const int crow = crow0 + r;
        if (crow < M) C[(size_t)crow * H + col] = acc[t][r] + badd;
      }
    }
  }
}

// ------------------------ self loop + edge scatter -------------------------
__global__ void self_init_kernel(const float* __restrict__ dinv, const float* __restrict__ ht,
                                 float* __restrict__ acc, int N) {
  size_t i = (size_t)blockIdx.x * blockDim.x + threadIdx.x;
  if (i < (size_t)N * H) {
    int row = (int)(i >> 7);
    float di = dinv[row];
    acc[i] = di * di * ht[i];
  }
}

__global__ __launch_bounds__(256)
void edge_scatter_kernel(const int* __restrict__ src, const int* __restrict__ dst,
                         const float* __restrict__ dinv, const float* __restrict__ ht,
                         float* __restrict__ acc, int E) {
  int e = blockIdx.x * 8 + (threadIdx.x >> 5);  // one wave per edge
  if (e >= E) return;
  int lane = threadIdx.x & 31;
  int s = src[e], d = dst[e];
  float norm = dinv[s] * dinv[d];
  const float4* hp = reinterpret_cast<const float4*>(ht + (size_t)s * H);
  float4 v = hp[lane];                          // 512B coalesced per wave
  float* ap = acc + (size_t)d * H + lane * 4;
  atomic_add_f32_global(ap + 0, norm * v.x);
  atomic_add_f32_global(ap + 1, norm * v.y);
  atomic_add_f32_global(ap + 2, norm * v.z);
  atomic_add_f32_global(ap + 3, norm * v.w);
}

// -------------------- bias + LayerNorm + ReLU + residual -------------------
__global__ __launch_bounds__(256)
void ln_relu_res_kernel(const float* __restrict__ acc, const float* __restrict__ bconv,
                        const float* __restrict__ g, const float* __restrict__ b,
                        const float* __restrict__ resid, float* __restrict__ out, int N) {
  int row = blockIdx.x * 8 + (threadIdx.x >> 5);  // one wave per node row
  if (row >= N) return;
  int lane = threadIdx.x & 31;

  const float4* ap = reinterpret_cast<const float4*>(acc + (size_t)row * H);
  float4 v  = ap[lane];
  float4 bb = reinterpret_cast<const float4*>(bconv)[lane];
  v.x += bb.x; v.y += bb.y; v.z += bb.z; v.w += bb.w;

  float s = v.x + v.y + v.z + v.w;
#pragma unroll
  for (int off = 16; off; off >>= 1) s += __shfl_xor(s, off, 32);
  float mu = s * (1.0f / 128.0f);

  float dx = v.x - mu, dy = v.y - mu, dz = v.z - mu, dw = v.w - mu;
  float q = dx * dx + dy * dy + dz * dz + dw * dw;
#pragma unroll
  for (int off = 16; off; off >>= 1) q += __shfl_xor(q, off, 32);
  float rs = rsqrtf(q * (1.0f / 128.0f) + 1e-5f);

  float4 gg = reinterpret_cast<const float4*>(g)[lane];
  float4 bl = reinterpret_cast<const float4*>(b)[lane];
  float4 rr = reinterpret_cast<const float4*>(resid + (size_t)row * H)[lane];

  float4 o;
  o.x = fmaxf(dx * rs * gg.x + bl.x, 0.0f) + rr.x;
  o.y = fmaxf(dy * rs * gg.y + bl.y, 0.0f) + rr.y;
  o.z = fmaxf(dz * rs * gg.z + bl.z, 0.0f) + rr.z;
  o.w = fmaxf(dw * rs * gg.w + bl.w, 0.0f) + rr.w;
  reinterpret_cast<float4*>(out + (size_t)row * H)[lane] = o;
}

// ------------------------------ mean pool ----------------------------------
__global__ void zero_kernel(float* __restrict__ p, int n) {
  int i = blockIdx.x * blockDim.x + threadIdx.x;
  if (i < n) p[i] = 0.0f;
}

__global__ __launch_bounds__(128)
void pool_sum_kernel(const float* __restrict__ h, float* __restrict__ pooled, int N) {
  int col = threadIdx.x;  // 128 threads: one column each
  float s = 0.0f;
  for (int r = blockIdx.x; r < N; r += gridDim.x) s += h[(size_t)r * H + col];
  atomic_add_f32_global(pooled + col, s);
}

// --------------------------------- head ------------------------------------
__global__ __launch_bounds__(256)
void head_kernel(const float* __restrict__ pooled,
                 const float* __restrict__ W1, const float* __restrict__ b1,
                 const float* __restrict__ g, const float* __restrict__ b,
                 const float* __restrict__ W2, const float* __restrict__ b2,
                 float* __restrict__ out, int N, int OUTN) {
  __shared__ float hm[H];
  __shared__ float zn[H];
  __shared__ float stats[2];
  int t = threadIdx.x;

  if (t < H) hm[t] = pooled[t] * (1.0f / (float)N);
  __syncthreads();

  if (t < H) {
    float s = b1[t];
    for (int k = 0; k < H; ++k) s = fmaf(hm[k], W1[k * H + t], s);
    zn[t] = s;
  }
  __syncthreads();

  if (t == 0) {
    float mu = 0.0f;
    for (int k = 0; k < H; ++k) mu += zn[k];
    mu *= (1.0f / 128.0f);
    float var = 0.0f;
    for (int k = 0; k < H; ++k) { float d = zn[k] - mu; var += d * d; }
    var *= (1.0f / 128.0f);
    stats[0] = mu;
    stats[1] = rsqrtf(var + 1e-5f);
  }
  __syncthreads();

  if (t < H) zn[t] = fmaxf((zn[t] - stats[0]) * stats[1] * g[t] + b[t], 0.0f);
  __syncthreads();

  if (t < OUTN) {
    float s = b2[t];
    for (int k = 0; k < H; ++k) s = fmaf(zn[k], W2[k * OUTN + t], s);
    out[t] = s;
  }
}

// ------------------------------- launcher ----------------------------------
extern "C" void kernel_launch(void* const* d_in, const int* in_sizes, int n_in,
                              void* d_out, int out_size, void* d_ws, size_t ws_size,
                              hipStream_t stream) {
  const float* x      = (const float*)d_in[0];
  const int*   ei     = (const int*)d_in[1];
  const float* W_emb  = (const float*)d_in[2];
  const float* b_emb  = (const float*)d_in[3];
  const float* W_conv = (const float*)d_in[4];   // [4,128,128]
  const float* b_conv = (const float*)d_in[5];   // [4,128]
  const float* ln_g   = (const float*)d_in[6];
  const float* ln_b   = (const float*)d_in[7];
  const float* W_res  = (const float*)d_in[8];   // [2,128,128]
  const float* b_res  = (const float*)d_in[9];
  const float* W_fc1  = (const float*)d_in[10];
  const float* b_fc1  = (const float*)d_in[11];
  const float* fcn_g  = (const float*)d_in[12];
  const float* fcn_b  = (const float*)d_in[13];
  const float* W_fc2  = (const float*)d_in[14];  // [128,200]
  const float* b_fc2  = (const float*)d_in[15];
  float* out = (float*)d_out;

  const int N = in_sizes[0] / 32;
  const int E = in_sizes[1] / 2;
  const int OUTN = out_size;
  const int* src = ei;
  const int* dst = ei + E;

  // workspace layout (floats)
  const size_t NH = (size_t)N * H;
  float* ws   = (float*)d_ws;
  float* bufA = ws;             // h (ping)
  float* bufB = bufA + NH;      // ht / h (pong)
  float* acc  = bufB + NH;      // aggregation accumulator
  float* res  = acc + NH;       // residual GEMM output (layer 2)
  float* deg  = res + NH;       // [N]
  float* dinv = deg + N;        // [N]
  float* pooled = dinv + N;     // [128]

  // 1) GCN normalization
  init_deg_kernel<<<(N + 255) / 256, 256, 0, stream>>>(deg, N);
  deg_accum_kernel<<<(E + 255) / 256, 256, 0, stream>>>(dst, deg, E);
  dinv_kernel<<<(N + 255) / 256, 256, 0, stream>>>(deg, dinv, N);

  const int gemm_blocks = (N + 127) / 128;

  // 2) embedding: h = x @ W_emb + b_emb   (K=32)
  gemm_wmma_kernel<32><<<gemm_blocks, 256, 0, stream>>>(x, W_emb, b_emb, bufA, N);

  float* cur = bufA;
  float* alt = bufB;

  const int selfBlocks = (int)((NH + 255) / 256);
  const int edgeBlocks = (E + 7) / 8;
  const int rowBlocks  = (N + 7) / 8;

  // 3) four GCN layers
  for (int i = 0; i < NLAYERS; ++i) {
    const float* resid = cur;
    if (i == 2) {  // residual = h @ W_res[0] + b_res[0]
      gemm_wmma_kernel<H><<<gemm_blocks, 256, 0, stream>>>(cur, W_res, b_res, res, N);
      resid = res;
    }
    // ht = h @ W_conv[i]   (bias folded in later)
    gemm_wmma_kernel<H><<<gemm_blocks, 256, 0, stream>>>(
        cur, W_conv + (size_t)i * H * H, nullptr, alt, N);
    // acc = dinv^2 * ht (self loops), then scatter edges
    self_init_kernel<<<selfBlocks, 256, 0, stream>>>(dinv, alt, acc, N);
    edge_scatter_kernel<<<edgeBlocks, 256, 0, stream>>>(src, dst, dinv, alt, acc, E);
    // h_next = relu(LN(acc + b_conv)) + residual   (written over ht buffer)
    ln_relu_res_kernel<<<rowBlocks, 256, 0, stream>>>(
        acc, b_conv + (size_t)i * H, ln_g + (size_t)i * H, ln_b + (size_t)i * H,
        resid, alt, N);
    float* tmp = cur; cur = alt; alt = tmp;
  }

  // 4) global mean pool + MLP head
  zero_kernel<<<1, 128, 0, stream>>>(pooled, H);
  pool_sum_kernel<<<256, 128, 0, stream>>>(cur, pooled, N);
  head_kernel<<<1, 256, 0, stream>>>(pooled, W_fc1, b_fc1, fcn_g, fcn_b,
                                     W_fc2, b_fc2, out, N, OUTN);
}